// KPlane_69423851372725
// MI455X (gfx1250) — compile-verified
//
#include <hip/hip_runtime.h>

typedef float v2f __attribute__((ext_vector_type(2)));
typedef float v8f __attribute__((ext_vector_type(8)));

namespace {

constexpr int kC        = 32;   // plane channels (= GEMM1 K)
constexpr int kH1       = 32;   // hidden width   (= GEMM1 N)
constexpr int kBlock    = 256;  // 8 wave32s
constexpr int kRowStr   = 33;   // padded LDS row stride (odd -> bank-conflict-free columns)

struct Bilin {
  int   o00, o01, o10, o11;
  float w00, w01, w10, w11;
};

__device__ __forceinline__ Bilin bilin_setup(float gx, float gy, int W, int H) {
  // align_corners=True convention, matches the reference
  float ix  = (gx + 1.0f) * 0.5f * (float)(W - 1);
  float iy  = (gy + 1.0f) * 0.5f * (float)(H - 1);
  float x0f = floorf(ix);
  float y0f = floorf(iy);
  float wx  = ix - x0f;
  float wy  = iy - y0f;
  int x0 = (int)x0f, y0 = (int)y0f;
  x0 = min(max(x0, 0), W - 1);
  y0 = min(max(y0, 0), H - 1);
  int x1 = min(x0 + 1, W - 1);
  int y1 = min(y0 + 1, H - 1);
  Bilin b;
  b.o00 = y0 * W + x0;  b.o01 = y0 * W + x1;
  b.o10 = y1 * W + x0;  b.o11 = y1 * W + x1;
  float iwx = 1.0f - wx, iwy = 1.0f - wy;
  b.w00 = iwx * iwy;  b.w01 = wx * iwy;
  b.w10 = iwx * wy;   b.w11 = wx * wy;
  return b;
}

__device__ __forceinline__ float bilin_fetch(const float* __restrict__ p,
                                             size_t cofs, const Bilin& b) {
  return b.w00 * p[cofs + b.o00] + b.w01 * p[cofs + b.o01] +
         b.w10 * p[cofs + b.o10] + b.w11 * p[cofs + b.o11];
}

} // namespace

__global__ __launch_bounds__(kBlock)
void triplane_mlp_kernel(const float* __restrict__ pts,
                         const float* __restrict__ plane_yx,
                         const float* __restrict__ plane_zx,
                         const float* __restrict__ plane_zy,
                         const float* __restrict__ w1,
                         const float* __restrict__ b1,
                         const float* __restrict__ w2,
                         const float* __restrict__ b2,
                         float* __restrict__ out, int N,
                         int Wyx, int Hyx, int Wzx, int Hzx, int Wzy, int Hzy)
{
  __shared__ float sFeat[kBlock * kRowStr];

  const int t      = (int)threadIdx.x;
  const int lane   = t & 31;
  const int wave   = t >> 5;
  const int col    = lane & 15;   // WMMA N / M-within-tile index
  const int laneHi = lane >> 4;   // upper 16 lanes flag

  int p  = (int)blockIdx.x * kBlock + t;
  int pc = min(p, N - 1);         // clamp so every lane does real work; EXEC stays all-1 for WMMA

  const float px = pts[(size_t)pc * 3 + 0];
  const float py = pts[(size_t)pc * 3 + 1];
  const float pz = pts[(size_t)pc * 3 + 2];

  const Bilin byx = bilin_setup(px, py, Wyx, Hyx);
  const Bilin bzx = bilin_setup(px, pz, Wzx, Hzx);
  const Bilin bzy = bilin_setup(py, pz, Wzy, Hzy);

  const size_t HW0 = (size_t)Wyx * (size_t)Hyx;
  const size_t HW1 = (size_t)Wzx * (size_t)Hzx;
  const size_t HW2 = (size_t)Wzy * (size_t)Hzy;

  // ---- Phase 1: gather tri-plane features, product over planes -> LDS row ----
  float* myRow = &sFeat[t * kRowStr];
#pragma unroll 4
  for (int c = 0; c < kC; ++c) {
    float s0 = bilin_fetch(plane_yx, (size_t)c * HW0, byx);
    float s1 = bilin_fetch(plane_zx, (size_t)c * HW1, bzx);
    float s2 = bilin_fetch(plane_zy, (size_t)c * HW2, bzy);
    myRow[c] = s0 * s1 * s2;
  }

  // ---- Preload w1 B-fragments (K=32 x N=32, f32 WMMA 16x16x4 layout) ----
  // B layout: lanes 0-15 -> K = k0+{0,1} in V0/V1; lanes 16-31 -> K = k0+{2,3}; N = lane%16.
  v2f bf[8][2];
#pragma unroll
  for (int ks = 0; ks < 8; ++ks) {
    const int kb = ks * 4 + laneHi * 2;
#pragma unroll
    for (int h = 0; h < 2; ++h) {
      bf[ks][h].x = w1[(kb + 0) * kH1 + h * 16 + col];
      bf[ks][h].y = w1[(kb + 1) * kH1 + h * 16 + col];
    }
  }
  const float bias0 = b1[col];
  const float bias1 = b1[16 + col];

  __syncthreads();

  // ---- Phase 2: GEMM1 (features[16x32] @ w1[32x32]) per 16-point tile via WMMA ----
#pragma unroll
  for (int tile = 0; tile < 2; ++tile) {
    const int rowBase = wave * 32 + tile * 16;
    v8f acc0 = {0.f, 0.f, 0.f, 0.f, 0.f, 0.f, 0.f, 0.f};
    v8f acc1 = {0.f, 0.f, 0.f, 0.f, 0.f, 0.f, 0.f, 0.f};
    // A layout: lanes 0-15 -> M=lane, V0/V1 = K0/K1; lanes 16-31 -> M=lane-16, V0/V1 = K2/K3.
    const float* aRow = &sFeat[(rowBase + col) * kRowStr + laneHi * 2];
#pragma unroll
    for (int ks = 0; ks < 8; ++ks) {
      v2f a;
      a.x = aRow[ks * 4 + 0];
      a.y = aRow[ks * 4 + 1];
      acc0 = __builtin_amdgcn_wmma_f32_16x16x4_f32(false, a, false, bf[ks][0],
                                                   (short)0, acc0, false, false);
      acc1 = __builtin_amdgcn_wmma_f32_16x16x4_f32(false, a, false, bf[ks][1],
                                                   (short)0, acc1, false, false);
    }
    // bias + ReLU, write h back in place over this wave's own feat rows
    // D layout: VGPR j -> M = j + (laneHi ? 8 : 0); N = lane%16.
#pragma unroll
    for (int j = 0; j < 8; ++j) {
      const int row = rowBase + j + laneHi * 8;
      const float h0 = fmaxf(acc0[j] + bias0, 0.0f);
      const float h1 = fmaxf(acc1[j] + bias1, 0.0f);
      sFeat[row * kRowStr + col]      = h0;
      sFeat[row * kRowStr + 16 + col] = h1;
    }
  }

  __syncthreads();

  // ---- Phase 3: GEMM2 (h[32] @ w2[32x4] + b2), scalar per point ----
  float o0 = b2[0], o1 = b2[1], o2 = b2[2], o3 = b2[3];
#pragma unroll
  for (int c = 0; c < kH1; ++c) {
    const float hv = myRow[c];
    o0 = fmaf(hv, w2[c * 4 + 0], o0);
    o1 = fmaf(hv, w2[c * 4 + 1], o1);
    o2 = fmaf(hv, w2[c * 4 + 2], o2);
    o3 = fmaf(hv, w2[c * 4 + 3], o3);
  }
  if (p < N) {
    *reinterpret_cast<float4*>(out + (size_t)p * 4) = make_float4(o0, o1, o2, o3);
  }
}

extern "C" void kernel_launch(void* const* d_in, const int* in_sizes, int n_in,
                              void* d_out, int out_size, void* d_ws, size_t ws_size,
                              hipStream_t stream) {
  (void)n_in; (void)out_size; (void)d_ws; (void)ws_size;
  const float* pts      = (const float*)d_in[0];
  const float* plane_yx = (const float*)d_in[1];
  const float* plane_zx = (const float*)d_in[2];
  const float* plane_zy = (const float*)d_in[3];
  const float* w1       = (const float*)d_in[4];
  const float* b1       = (const float*)d_in[5];
  const float* w2       = (const float*)d_in[6];
  const float* b2       = (const float*)d_in[7];
  float* out = (float*)d_out;

  const int N = in_sizes[0] / 3;
  // res = (727, 314, 1300); planes are (C, H, W):
  // plane_yx: W=727 (x), H=314 (y); plane_zx: W=727 (x), H=1300 (z); plane_zy: W=314 (y), H=1300 (z)
  const int Wyx = 727, Hyx = 314;
  const int Wzx = 727, Hzx = 1300;
  const int Wzy = 314, Hzy = 1300;

  const int blocks = (N + kBlock - 1) / kBlock;
  triplane_mlp_kernel<<<blocks, kBlock, 0, stream>>>(
      pts, plane_yx, plane_zx, plane_zy, w1, b1, w2, b2, out, N,
      Wyx, Hyx, Wzx, Hzx, Wzy, Hzy);
}